// LocalContrastiveLoss_49684181680812
// MI455X (gfx1250) — compile-verified
//
#include <hip/hip_runtime.h>
#include <math.h>

// Problem constants (match reference setup_inputs)
#define BB   8
#define CC   256
#define NPix 16384      // 128*128
#define NCls 4
#define PP   3
#define KK   3
#define TINV 10.0f      // 1/T, T=0.1
#define LAMF 0.1f

// Workspace byte offsets (total ~756KB)
#define OFF_COUNTS 0u         // 32 int
#define OFF_SIDX   128u       // 480 int   (ends at 2048)
#define OFF_A      2048u      // 192*256 f32 (anchors: pos rows 0..95, pos1 rows 96..191)
#define OFF_BT     198656u    // 256*256 f32 (targets^T: mu1 0..31, mu2 32..63, neg 64..159, neg1 160..255)
#define OFF_TMP    460800u    // 96*256 f32 (normalized draw-3 samples, feed mu2)
#define OFF_D      559104u    // 192*256 f32 (score matrix)

typedef float v2f __attribute__((ext_vector_type(2)));
typedef float v8f __attribute__((ext_vector_type(8)));

// ---------------- deterministic hash-uniform (stand-in for jax threefry) ------------
__device__ __forceinline__ float hash_uniform(unsigned draw, unsigned b, unsigned c, unsigned i) {
    unsigned long long x =
        ((unsigned long long)(draw * 0x9E3779B1u ^ b * 0x85EBCA77u ^ c * 0xC2B2AE3Du) << 32) |
        (unsigned long long)i;
    x += 0x9E3779B97F4A7C15ull;
    x = (x ^ (x >> 30)) * 0xBF58476D1CE4E5B9ull;
    x = (x ^ (x >> 27)) * 0x94D049BB133111EBull;
    x = x ^ (x >> 31);
    return (float)(unsigned)(x >> 40) * (1.0f / 16777216.0f);   // [0,1)
}

// ---------------- kernel 1: per-(b,class,draw) top-3 random pixel selection + counts ----
// grid = 5*B*NC = 160 blocks, 256 threads. Reads only labels (512KB total across grid).
__global__ void k_sample(const int* __restrict__ labels,
                         int* __restrict__ counts,
                         int* __restrict__ sidx) {
    const int bid  = blockIdx.x;
    const int draw = bid / (BB * NCls);
    const int b    = (bid / NCls) % BB;
    const int c    = bid % NCls;
    const int tid  = threadIdx.x;

    float tv0 = -1.f, tv1 = -1.f, tv2 = -1.f;
    int   ti0 = 0,    ti1 = 0,    ti2 = 0;
    int   cnt = 0;
    const int* lab = labels + b * NPix;
    for (int i = tid; i < NPix; i += 256) {
        if (lab[i] == c) {
            cnt++;
            float u = hash_uniform((unsigned)draw, (unsigned)b, (unsigned)c, (unsigned)i);
            if (u > tv2) {
                if (u > tv0)      { tv2 = tv1; ti2 = ti1; tv1 = tv0; ti1 = ti0; tv0 = u; ti0 = i; }
                else if (u > tv1) { tv2 = tv1; ti2 = ti1; tv1 = u;   ti1 = i; }
                else              { tv2 = u;   ti2 = i; }
            }
        }
    }

    __shared__ float sv[256 * 3];
    __shared__ int   si[256 * 3];
    __shared__ int   sc[256];
    sv[tid*3+0] = tv0; sv[tid*3+1] = tv1; sv[tid*3+2] = tv2;
    si[tid*3+0] = ti0; si[tid*3+1] = ti1; si[tid*3+2] = ti2;
    sc[tid] = cnt;
    __syncthreads();

    for (int s = 128; s > 0; s >>= 1) {
        if (tid < s) {
            sc[tid] += sc[tid + s];
            float cv[6]; int ci[6];
            #pragma unroll
            for (int q = 0; q < 3; q++) {
                cv[q]     = sv[tid*3 + q];       ci[q]     = si[tid*3 + q];
                cv[q + 3] = sv[(tid+s)*3 + q];   ci[q + 3] = si[(tid+s)*3 + q];
            }
            #pragma unroll
            for (int p = 0; p < 3; p++) {       // partial selection sort: top-3 of 6
                int m = p;
                #pragma unroll
                for (int q = p + 1; q < 6; q++) if (cv[q] > cv[m]) m = q;
                float fv = cv[m]; int fi = ci[m];
                cv[m] = cv[p]; ci[m] = ci[p];
                cv[p] = fv;    ci[p] = fi;
                sv[tid*3 + p] = cv[p];
                si[tid*3 + p] = ci[p];
            }
        }
        __syncthreads();
    }
    if (tid == 0) {
        if (draw == 0) counts[b * NCls + c] = sc[0];
        const int base = ((draw * BB + b) * NCls + c) * 3;
        sidx[base + 0] = si[0];
        sidx[base + 1] = si[1];
        sidx[base + 2] = si[2];
    }
}

// ---------------- kernel 2: gather sampled vectors + L2 normalize (eps=1e-12) ---------
// grid = 480 blocks (one wave per vector), 32 threads.
__global__ void k_gather(const float* __restrict__ feat, const int* __restrict__ sidx,
                         float* __restrict__ Am, float* __restrict__ Bt,
                         float* __restrict__ Tm) {
    const int vb   = blockIdx.x;           // draw*96 + (b*4+c)*3 + j
    const int lane = threadIdx.x;
    const int draw = vb / 96;
    const int rem  = vb % 96;
    const int b    = rem / (NCls * PP);
    const int pixel = sidx[vb];

    const float* src = feat + (size_t)b * (CC * NPix) + pixel;   // channel stride = NPix
    __builtin_prefetch(src, 0, 3);
    float v[8];
    float ss = 0.f;
    #pragma unroll
    for (int k = 0; k < 8; k++) {
        v[k] = src[(size_t)(lane + 32 * k) * NPix];
        ss += v[k] * v[k];
    }
    #pragma unroll
    for (int m = 16; m > 0; m >>= 1) ss += __shfl_xor(ss, m, 32);
    const float scale = 1.0f / fmaxf(sqrtf(ss), 1e-12f);

    float* dst;
    if      (draw == 0) dst = Am + (size_t)rem * CC;          // pos
    else if (draw == 2) dst = Am + (size_t)(96 + rem) * CC;   // pos1
    else if (draw == 1) dst = Bt + (size_t)(64 + rem) * CC;   // neg
    else if (draw == 4) dst = Bt + (size_t)(160 + rem) * CC;  // neg1
    else                dst = Tm + (size_t)rem * CC;          // draw 3 -> mu2 source
    #pragma unroll
    for (int k = 0; k < 8; k++) dst[lane + 32 * k] = v[k] * scale;
}

// ---------------- kernel 2b: mu = unit(mean of 3 normalized vectors), eps=1e-8 --------
// grid = 64 blocks (mu1 for 32 (b,c) groups, then mu2), 32 threads.
__global__ void k_mu(const float* __restrict__ Am, const float* __restrict__ Tm,
                     float* __restrict__ Bt) {
    const int which = blockIdx.x >> 5;     // 0 = mu1 (from pos rows), 1 = mu2 (from Tm)
    const int g     = blockIdx.x & 31;     // b*4 + c
    const int lane  = threadIdx.x;
    const float* s  = (which == 0 ? Am : Tm) + (size_t)g * 3 * CC;
    float v[8];
    float ss = 0.f;
    #pragma unroll
    for (int k = 0; k < 8; k++) {
        const int idx = lane + 32 * k;
        v[k] = (s[idx] + s[CC + idx] + s[2 * CC + idx]) * (1.0f / 3.0f);
        ss += v[k] * v[k];
    }
    #pragma unroll
    for (int m = 16; m > 0; m >>= 1) ss += __shfl_xor(ss, m, 32);
    const float scale = 1.0f / fmaxf(sqrtf(ss), 1e-8f);
    float* dst = Bt + (size_t)(which * 32 + g) * CC;
    #pragma unroll
    for (int k = 0; k < 8; k++) dst[lane + 32 * k] = v[k] * scale;
}

// ---------------- kernel 3: WMMA GEMM  D(192x256) = A(192x256) * B^T(256x256) ---------
// One wave per 16x16 tile, K=256 via 64 chained v_wmma_f32_16x16x4_f32 (exact f32 math).
// Tiles staged into LDS with the gfx1250 async-copy path (global_load_async_to_lds_b128,
// tracked on ASYNCcnt), bypassing the VGPR file. LDS rows padded to 260 floats so a
// K-slice read hits 16 distinct banks.
#define LR 260
__global__ void k_wmma(const float* __restrict__ Am, const float* __restrict__ Bt,
                       float* __restrict__ D) {
    __shared__ float As[16 * LR];
    __shared__ float Bs[16 * LR];
    const int lane = threadIdx.x;
    const int tm   = blockIdx.x >> 4;      // 12 row tiles
    const int tn   = blockIdx.x & 15;      // 16 col tiles

    // --- async stage both 16x256 f32 tiles into LDS (16B per lane per op) ---
    const unsigned long long gA = (unsigned long long)(uintptr_t)(Am + (size_t)tm * 16 * CC);
    const unsigned long long gB = (unsigned long long)(uintptr_t)(Bt + (size_t)tn * 16 * CC);
    // low 32 bits of a generic LDS pointer == LDS byte offset (shared aperture layout)
    const unsigned lAs = (unsigned)(uintptr_t)&As[0];
    const unsigned lBs = (unsigned)(uintptr_t)&Bs[0];
    #pragma unroll 4
    for (int it = 0; it < 32; ++it) {                 // 1024 float4 / 32 lanes
        const int i = it * 32 + lane;
        const int r = i >> 6, o = i & 63;             // row, float4-within-row
        const unsigned ldsOff = (unsigned)(r * (LR * 4) + o * 16);
        const unsigned glbOff = (unsigned)(r * (CC * 4) + o * 16);
        const unsigned la = lAs + ldsOff;
        const unsigned lb = lBs + ldsOff;
        const unsigned long long ga = gA + glbOff;
        const unsigned long long gb = gB + glbOff;
        asm volatile("global_load_async_to_lds_b128 %0, %1, off"
                     :: "v"(la), "v"(ga) : "memory");
        asm volatile("global_load_async_to_lds_b128 %0, %1, off"
                     :: "v"(lb), "v"(gb) : "memory");
    }
    asm volatile("s_wait_asynccnt 0" ::: "memory");   // all async copies landed in LDS
    __syncthreads();

    v8f acc = {};
    const int row  = lane & 15;            // A: M row / B: N column
    const int koff = (lane >> 4) * 2;      // lanes 16..31 hold K+2,K+3
    const float* ap = As + row * LR + koff;
    const float* bp = Bs + row * LR + koff;
    #pragma unroll
    for (int k0 = 0; k0 < CC; k0 += 4) {
        v2f a; a.x = ap[k0]; a.y = ap[k0 + 1];
        v2f bfr; bfr.x = bp[k0]; bfr.y = bp[k0 + 1];
        acc = __builtin_amdgcn_wmma_f32_16x16x4_f32(false, a, false, bfr,
                                                    (short)0, acc, false, false);
    }

    // 16x16 f32 C/D layout: VGPR v -> M = v + 8*(lane>=16), N = lane&15
    const int rhalf = (lane >> 4) * 8;
    const int col   = tn * 16 + (lane & 15);
    #pragma unroll
    for (int v = 0; v < 8; v++) {
        const int rr = tm * 16 + rhalf + v;
        D[rr * 256 + col] = acc[v];
    }
}

// ---------------- kernel 4: masked softmax-contrast terms + scalar reduction ----------
__global__ void k_loss(const float* __restrict__ D, const int* __restrict__ counts,
                       float* __restrict__ out) {
    __shared__ int   cnt[32];
    __shared__ float red[128];
    const int t = threadIdx.x;
    if (t < 32) cnt[t] = counts[t];
    __syncthreads();

    float loss = 0.f;
    if (t < 96) {                                    // t = (b*4+c)*3 + p
        const int b = t / 12, c = (t / 3) & 3;
        bool has_neg = false;
        #pragma unroll
        for (int e = 0; e < NCls; e++)
            if (e != c && cnt[b * 4 + e] > 0) has_neg = true;
        const bool enough = cnt[b * 4 + c] >= PP;

        // part 1: within-image
        const float* d1 = D + (size_t)t * 256;
        const float sp1 = expf(d1[b * 4 + c] * TINV);       // col = mu1[b,c]
        float sn1 = 0.f;
        #pragma unroll
        for (int e = 0; e < NCls; e++) if (e != c) {
            const int kv = min(cnt[b * 4 + e], KK);
            for (int k = 0; k < kv; k++)
                sn1 += expf(d1[64 + (b * 4 + e) * 3 + k] * TINV);
        }
        if (enough && has_neg) loss += -logf(sp1 / (sp1 + sn1 + 1e-8f));

        // part 2: cross-image
        const float* d2 = D + (size_t)(96 + t) * 256;
        float sn2 = 0.f;
        #pragma unroll
        for (int e = 0; e < NCls; e++) if (e != c) {
            const int kv = min(cnt[b * 4 + e], KK);
            for (int k = 0; k < kv; k++)
                sn2 += expf(d2[160 + (b * 4 + e) * 3 + k] * TINV);
        }
        if (enough && has_neg) {
            for (int a = 0; a < BB; a++) if (a != b && cnt[a * 4 + c] >= PP) {
                const float sp2 = expf(d2[32 + a * 4 + c] * TINV);   // col = mu2[a,c]
                loss += -logf(sp2 / (sp2 + sn2 + 1e-8f));
            }
        }
    }
    red[t] = loss;
    __syncthreads();
    for (int s = 64; s > 0; s >>= 1) {
        if (t < s) red[t] += red[t + s];
        __syncthreads();
    }
    if (t == 0) out[0] = LAMF * red[0] / ((float)(BB * PP) + 1e-8f);
}

extern "C" void kernel_launch(void* const* d_in, const int* in_sizes, int n_in,
                              void* d_out, int out_size, void* d_ws, size_t ws_size,
                              hipStream_t stream) {
    const float* features = (const float*)d_in[0];
    const int*   labels   = (const int*)d_in[1];
    char* ws = (char*)d_ws;
    int*   counts = (int*)(ws + OFF_COUNTS);
    int*   sidx   = (int*)(ws + OFF_SIDX);
    float* Am     = (float*)(ws + OFF_A);
    float* Bt     = (float*)(ws + OFF_BT);
    float* Tm     = (float*)(ws + OFF_TMP);
    float* D      = (float*)(ws + OFF_D);

    hipLaunchKernelGGL(k_sample, dim3(5 * BB * NCls), dim3(256), 0, stream, labels, counts, sidx);
    hipLaunchKernelGGL(k_gather, dim3(480), dim3(32), 0, stream, features, sidx, Am, Bt, Tm);
    hipLaunchKernelGGL(k_mu,     dim3(64),  dim3(32), 0, stream, Am, Tm, Bt);
    hipLaunchKernelGGL(k_wmma,   dim3(192), dim3(32), 0, stream, Am, Bt, D);
    hipLaunchKernelGGL(k_loss,   dim3(1),   dim3(128), 0, stream, D, counts, (float*)d_out);
}